// EffectiveProbability_79113297592736
// MI455X (gfx1250) — compile-verified
//
#include <hip/hip_runtime.h>

// ---------------------------------------------------------------------------
// Bayes fusion posterior for semantic segmentation (MI455X / gfx1250).
//
//   post[pix, i] ∝ prior[pix, i] * sum_j CM[i, j] * current[pix, j]
//
// Row-normalizations of prior/current are algebraically redundant (absorbed
// by the final per-pixel normalization), so only the final sum/divide is done.
//
// Core matvec batched as V_WMMA_F32_16X16X4_F32 (fp32-exact):
//   D[class][pixel] = A(CM: M=class,K) x B(current: K,N=pixel) + D
// 21 classes -> 2 M-tiles, 21-deep K -> 6 chunks of 4 (zero padded).
// Memory-bound: ~528 MB one-pass stream -> NT hints, 32-bit saddr+voffset
// addressing, LDS-staged 128B-coalesced output.
// ---------------------------------------------------------------------------

typedef __attribute__((ext_vector_type(2))) float v2f;
typedef __attribute__((ext_vector_type(8))) float v8f;

#define CCH   21
#define SPIX  (512 * 512)          // pixels per image (1 << 18)
#define NPIX  (8 * SPIX)           // total pixels
#define ITERS 4                    // 32-pixel groups per wave
#define WVB   8                    // waves per block (256 threads)

// Exchange lanes L <-> L^16 across the wave32: ds_swizzle SWAPX16
// (group-of-32 encoding: offset[14:10]=xor=0x10, offset[4:0]=and=0x1f).
__device__ __forceinline__ float swap_half(float x) {
  return __int_as_float(__builtin_amdgcn_ds_swizzle(__float_as_int(x), 0x401f));
}

__global__ __launch_bounds__(256) void bayes_fuse_wmma(
    const float* __restrict__ prior,
    const float* __restrict__ current,
    const float* __restrict__ cm,
    float* __restrict__ out)
{
  __shared__ float stage[WVB * 32 * CCH];          // 672 floats per wave

  const int tid    = threadIdx.x;
  const int lane   = tid & 31;
  const int wlocal = tid >> 5;
  const int lo     = lane & 15;                    // N index (pixel) / M index (class)
  const int hi     = lane >> 4;                    // lane-half selects K/M offset

  // --- CM into A-operand VGPR layout, held live across the pixel loop. ---
  // aCM[ct][q][j] = CM[class][k], class = 16*ct + lo, k = 4*q + 2*hi + j
  float aCM[2][6][2];
#pragma unroll
  for (int ct = 0; ct < 2; ++ct) {
    const int m  = 16 * ct + lo;
    const int mc = (m < CCH) ? m : (CCH - 1);
#pragma unroll
    for (int q = 0; q < 6; ++q) {
#pragma unroll
      for (int j = 0; j < 2; ++j) {
        const int k  = 4 * q + 2 * hi + j;
        const int kc = (k < CCH) ? k : (CCH - 1);
        const float v = cm[mc * CCH + kc];
        aCM[ct][q][j] = (m < CCH && k < CCH) ? v : 0.0f;
      }
    }
  }

  const int gwave = blockIdx.x * WVB + wlocal;

  for (int it = 0; it < ITERS; ++it) {
    const int g0  = (gwave * ITERS + it) * 32;     // first of 32 contiguous pixels
    const int img = g0 >> 18;                      // image index (SPIX = 1<<18)
    const int pix = g0 & (SPIX - 1);

    // Uniform per-image SGPR bases; all per-lane offsets are 32-bit elements.
    const float* __restrict__ curI = current + (size_t)img * (CCH * SPIX);
    const float* __restrict__ priI = prior   + (size_t)img * (CCH * SPIX);

    // Prefetch next group's leading cachelines (global_prefetch_b8).
    if (it + 1 < ITERS) {
      const unsigned np = (unsigned)((g0 + 32) & (SPIX - 1)) + (unsigned)lane;
      __builtin_prefetch(curI + np, 0, 1);
      __builtin_prefetch(priI + np, 0, 1);
    }

    v8f D[2][2];                                   // [class-tile][pixel-tile]
    D[0][0] = (v8f){}; D[0][1] = (v8f){};
    D[1][0] = (v8f){}; D[1][1] = (v8f){};

    // ---- K loop: 6 chunks of 4, 4 WMMAs per chunk --------------------------
#pragma unroll
    for (int q = 0; q < 6; ++q) {
      const int  kA   = 4 * q + 2 * hi;            // this lane-half's K pair
      const bool tail = (4 * q + 3 >= CCH);        // compile-time: only q == 5
      // Clamped channel offsets (clamp needed only in the tail chunk).
      const unsigned off0 = (unsigned)(tail ? ((kA < CCH) ? kA : (CCH - 1)) : kA) * SPIX;
      const unsigned off1 = (unsigned)(kA + 1) * SPIX;   // unused in tail
#pragma unroll
      for (int pt = 0; pt < 2; ++pt) {
        const unsigned pgi = (unsigned)(pix + 16 * pt + lo);
        float bx = __builtin_nontemporal_load(curI + off0 + pgi);
        float by = 0.0f;
        if (!tail) {
          by = __builtin_nontemporal_load(curI + off1 + pgi);
        } else {
          bx = (kA < CCH) ? bx : 0.0f;             // hi==1 tail lanes are padding
        }
        v2f Bv; Bv.x = bx; Bv.y = by;
#pragma unroll
        for (int ct = 0; ct < 2; ++ct) {
          v2f Av; Av.x = aCM[ct][q][0]; Av.y = aCM[ct][q][1];
          D[ct][pt] = __builtin_amdgcn_wmma_f32_16x16x4_f32(
              false, Av, false, Bv, (short)0, D[ct][pt], false, false);
        }
      }
    }

    // ---- posterior: p * pred, per-pixel normalize, stage to LDS ------------
#pragma unroll
    for (int pt = 0; pt < 2; ++pt) {
      const unsigned pgi = (unsigned)(pix + 16 * pt + lo);
      float post0[8], post1[5];
      float acc = 0.0f;
#pragma unroll
      for (int r = 0; r < 8; ++r) {                // classes r + 8*hi (< 16)
        const unsigned cls = (unsigned)(r + 8 * hi);
        const float pv = __builtin_nontemporal_load(priI + cls * SPIX + pgi);
        const float po = pv * D[0][pt][r];
        post0[r] = po;
        acc += po;
      }
#pragma unroll
      for (int r = 0; r < 5; ++r) {                // classes 16..20 (hi==0); hi==1 -> D==0
        const int cls = 16 + r + 8 * hi;
        const unsigned clc = (unsigned)((cls < CCH) ? cls : (CCH - 1));
        const float pv = __builtin_nontemporal_load(priI + clc * SPIX + pgi);
        const float po = pv * D[1][pt][r];         // zero-padded A rows => D==0 when invalid
        post1[r] = po;
        acc += po;
      }
      // lane P holds classes {0..7, 16..20}; lane P+16 holds {8..15}; swap halves.
      const float total = acc + swap_half(acc);
      const float inv   = 1.0f / total;

      float* wl = &stage[(wlocal * 32 + 16 * pt + lo) * CCH];
#pragma unroll
      for (int r = 0; r < 8; ++r)
        wl[r + 8 * hi] = post0[r] * inv;
      if (hi == 0) {
#pragma unroll
        for (int r = 0; r < 5; ++r)
          wl[16 + r] = post1[r] * inv;
      }
    }

    __builtin_amdgcn_wave_barrier();               // same-wave LDS ordering point

    // ---- drain: 672 contiguous floats -> coalesced 128B NT stores ----------
    float* __restrict__ ob = out + (size_t)g0 * CCH;
    const float* rl = &stage[wlocal * 32 * CCH];
#pragma unroll
    for (int i = 0; i < CCH; ++i) {
      const float v = rl[lane + 32 * i];
      __builtin_nontemporal_store(v, ob + lane + 32 * i);
    }
  }
}

extern "C" void kernel_launch(void* const* d_in, const int* in_sizes, int n_in,
                              void* d_out, int out_size, void* d_ws, size_t ws_size,
                              hipStream_t stream) {
  const float* prior   = (const float*)d_in[0];
  const float* current = (const float*)d_in[1];
  const float* cm      = (const float*)d_in[2];
  float* out           = (float*)d_out;

  const int pixels_per_block = WVB * ITERS * 32;   // 1024
  dim3 grid(NPIX / pixels_per_block);              // 2048
  dim3 block(256);
  bayes_fuse_wmma<<<grid, block, 0, stream>>>(prior, current, cm, out);
}